// EquivariantBallAttention_1803886265810
// MI455X (gfx1250) — compile-verified
//
#include <hip/hip_runtime.h>
#include <math.h>

// ---------------------------------------------------------------------------
// EquivariantBallAttention for MI455X (gfx1250, wave32, WMMA)
//   N=65536, BALL=256 -> B=256 balls, H=8 heads, HID_MV=4, HID_S=8
//   qf/kf dim = 40, vf dim = 72 per head
// Workspace layout (floats):
//   qf : [B*H, 256, 40]   = 20971520
//   kf : [B*H, 256, 40]   = 20971520
//   vf : [B*H, 256, 72]   = 37748736
//   of : [B*H, 256, 72]   = 37748736
// All GEMM-shaped work (equi_linear projections, QK^T, P*V) runs on
// V_WMMA_F32_16X16X4_F32 (fp32 in/out -> bit-compatible-ish with reference).
// ---------------------------------------------------------------------------

typedef float v2f __attribute__((ext_vector_type(2)));
typedef float v8f __attribute__((ext_vector_type(8)));

#define NPTS   65536
#define NBALL  256
#define NHEAD  8
#define QKDIM  40
#define VDIM   72
#define VPAD   80

#define QF_ELEMS  (2048u * 256u * 40u)   // 20971520
#define VF_ELEMS  (2048u * 256u * 72u)   // 37748736

#define WMMA_F32(A, B, C) \
  __builtin_amdgcn_wmma_f32_16x16x4_f32(false, (A), false, (B), (short)0, (C), false, false)

// Sparse BASIS contraction as 24 (out_comp, in_comp, weight_k) contributions.
#define NCONTRIB 24
#define CONTRIB_TABLES                                                        \
  const int CAa[NCONTRIB] = {0, 1, 2, 3, 4, 5, 6, 7, 8, 9, 10, 11,           \
                             12, 13, 14, 15, 1, 5, 6, 7, 11, 12, 13, 15};    \
  const int CBb[NCONTRIB] = {0, 1, 2, 3, 4, 5, 6, 7, 8, 9, 10, 11,           \
                             12, 13, 14, 15, 0, 2, 3, 4, 8, 9, 10, 14};      \
  const int CKk[NCONTRIB] = {0, 1, 1, 1, 1, 2, 2, 2, 2, 2, 2, 3,             \
                             3, 3, 3, 4, 5, 6, 6, 6, 7, 7, 7, 8};

__device__ __forceinline__ v8f v8f_zero() {
  return (v8f){0.f, 0.f, 0.f, 0.f, 0.f, 0.f, 0.f, 0.f};
}

// ---------------------------------------------------------------------------
// Kernel A1 (WMMA): multivector QKV projection.
// One wave = one 16-point tile (M). N-tiles of 16 out-channels (6 of them).
// A fragments: lane L -> (M = L%16 point, K = in-channel i = 4c+2*(L/16)+{0,1})
// B fragments: lane L -> (K = in-channel,  N = out-channel o = ntile*16+L%16)
// D tile:      lane L -> (N = o = L%16, M = point = vgpr + 8*(L/16))
// grid = N/64 blocks x 128 threads (4 waves).
// ---------------------------------------------------------------------------
__global__ __launch_bounds__(128, 1)
void qkv_mv_wmma(const float* __restrict__ x_mv, const float* __restrict__ x_s,
                 const float* __restrict__ w_mv, const float* __restrict__ w_s2mv,
                 const float* __restrict__ b_mv,
                 float* __restrict__ qf, float* __restrict__ kf,
                 float* __restrict__ vf) {
  CONTRIB_TABLES
  int wv = threadIdx.x >> 5, lane = threadIdx.x & 31;
  int hl = lane >> 4, lm = lane & 15, koff = hl * 2;
  int n0 = (blockIdx.x * 4 + wv) * 16;

  // Preload A fragments for all 16 components (reused across 6 n-tiles).
  v2f xa[16][4];
  const float* xrow = x_mv + (size_t)(n0 + lm) * 256;
#pragma unroll
  for (int c = 0; c < 4; ++c) {
    int i0 = c * 4 + koff;
#pragma unroll
    for (int bc = 0; bc < 16; ++bc) {
      xa[bc][c].x = xrow[i0 * 16 + bc];
      xa[bc][c].y = xrow[(i0 + 1) * 16 + bc];
    }
  }
  // A fragments for the scalar->mv (component 0) term, K = 32.
  v2f xsf[8];
  const float* srow = x_s + (size_t)(n0 + lm) * 32;
#pragma unroll
  for (int c = 0; c < 8; ++c) {
    int s0 = c * 4 + koff;
    xsf[c].x = srow[s0];
    xsf[c].y = srow[s0 + 1];
  }

#pragma unroll 1
  for (int nt = 0; nt < 6; ++nt) {
    int o = nt * 16 + lm;
    const float* wbase = w_mv + (size_t)o * (16 * 9);  // + i*9 + k

    v8f acc[16];
#pragma unroll
    for (int a = 0; a < 16; ++a) acc[a] = v8f_zero();

#pragma unroll
    for (int ci = 0; ci < NCONTRIB; ++ci) {
      int a = CAa[ci], bc = CBb[ci], k = CKk[ci];
#pragma unroll
      for (int c = 0; c < 4; ++c) {
        int i0 = c * 4 + koff;
        v2f wb;
        wb.x = wbase[i0 * 9 + k];
        wb.y = wbase[(i0 + 1) * 9 + k];
        acc[a] = WMMA_F32(xa[bc][c], wb, acc[a]);
      }
    }
    // scalar->mv into component 0 (K = 32)
    const float* w2 = w_s2mv + (size_t)o * 32;
#pragma unroll
    for (int c = 0; c < 8; ++c) {
      int s0 = c * 4 + koff;
      v2f wb;
      wb.x = w2[s0];
      wb.y = w2[s0 + 1];
      acc[0] = WMMA_F32(xsf[c], wb, acc[0]);
    }
    float bo = b_mv[o];
#pragma unroll
    for (int vg = 0; vg < 8; ++vg) acc[0][vg] += bo;

    // scatter into qf/kf/vf;  o = ((t*8+h)*4+m)
    int t = nt >> 1;                  // uniform per wave
    int hh = (o >> 2) & 7, m = o & 3;
    if (t < 2) {
      const int INNER[8] = {0, 2, 3, 4, 8, 9, 10, 14};
      float* base = (t == 0 ? qf : kf);
#pragma unroll
      for (int vg = 0; vg < 8; ++vg) {
        int nn = n0 + vg + hl * 8;
        size_t row = ((size_t)((nn >> 8) * NHEAD + hh) * NBALL + (nn & 255));
        float* dst = base + row * QKDIM + m * 8;
#pragma unroll
        for (int j = 0; j < 8; ++j) dst[j] = acc[INNER[j]][vg];
      }
    } else {
#pragma unroll
      for (int vg = 0; vg < 8; ++vg) {
        int nn = n0 + vg + hl * 8;
        size_t row = ((size_t)((nn >> 8) * NHEAD + hh) * NBALL + (nn & 255));
        float* dst = vf + row * VDIM + m * 16;
#pragma unroll
        for (int c4 = 0; c4 < 4; ++c4) {
          float4 v;
          v.x = acc[c4 * 4 + 0][vg];
          v.y = acc[c4 * 4 + 1][vg];
          v.z = acc[c4 * 4 + 2][vg];
          v.w = acc[c4 * 4 + 3][vg];
          ((float4*)dst)[c4] = v;
        }
      }
    }
  }
}

// ---------------------------------------------------------------------------
// Kernel A2: scalar QKV projection + RoPE (small: ~0.6 GFMA, stays VALU).
// One thread per (point, t*8+h) computing a full 8-dim head vector.
// ---------------------------------------------------------------------------
__global__ __launch_bounds__(256)
void qkv_s_kernel(const float* __restrict__ x_mv, const float* __restrict__ x_s,
                  const float* __restrict__ w_s, const float* __restrict__ w_mv2s,
                  const float* __restrict__ b_s,
                  float* __restrict__ qf, float* __restrict__ kf,
                  float* __restrict__ vf) {
  __shared__ float sw[192 * 32];
  __shared__ float sm2[192 * 16];
  __shared__ float sb[192];
  for (int idx = threadIdx.x; idx < 192 * 32; idx += 256) sw[idx] = w_s[idx];
  for (int idx = threadIdx.x; idx < 192 * 16; idx += 256) sm2[idx] = w_mv2s[idx];
  if (threadIdx.x < 192) sb[threadIdx.x] = b_s[threadIdx.x];
  __syncthreads();

  int tid = blockIdx.x * 256 + threadIdx.x;
  int n = tid / 24, g = tid % 24;  // g = t*8+h
  int t = g >> 3, hh = g & 7;

  float acc[8];
#pragma unroll
  for (int j = 0; j < 8; ++j) acc[j] = sb[g * 8 + j];

  const float* xsn = x_s + (size_t)n * 32;
#pragma unroll 8
  for (int s = 0; s < 32; ++s) {
    float xs = xsn[s];
#pragma unroll
    for (int j = 0; j < 8; ++j) acc[j] += sw[(g * 8 + j) * 32 + s] * xs;
  }
  const float* xmn = x_mv + (size_t)n * 256;
#pragma unroll 4
  for (int c = 0; c < 16; ++c) {
    float xc = xmn[c * 16];  // component 0 of channel c
#pragma unroll
    for (int j = 0; j < 8; ++j) acc[j] += sm2[(g * 8 + j) * 16 + c] * xc;
  }

  int b = n >> 8, q = n & 255;
  if (t < 2) {
    // RoPE: d=8, half=4, base 4096 => freq_j = 2^(-3j)
#pragma unroll
    for (int j = 0; j < 4; ++j) {
      float freq = exp2f(-3.0f * (float)j);
      float ang = (float)q * freq;
      float co = cosf(ang), si = sinf(ang);
      float r1 = acc[j] * co - acc[j + 4] * si;
      float r2 = acc[j] * si + acc[j + 4] * co;
      acc[j] = r1;
      acc[j + 4] = r2;
    }
    float* dst = (t == 0 ? qf : kf) +
                 ((size_t)(b * NHEAD + hh) * NBALL + q) * QKDIM + 32;
#pragma unroll
    for (int j = 0; j < 8; ++j) dst[j] = acc[j];
  } else {
    float* dst = vf + ((size_t)(b * NHEAD + hh) * NBALL + q) * VDIM + 64;
#pragma unroll
    for (int j = 0; j < 8; ++j) dst[j] = acc[j];
  }
}

// ---------------------------------------------------------------------------
// Kernel B: attention per (ball, head) with V_WMMA_F32_16X16X4_F32.
// 128 threads = 4 wave32; each wave handles 4 query tiles of 16 rows.
// Dynamic LDS: kf 40KB + vf(pad 80) 80KB + pos 4KB + P-staging 4KB = 128KB.
// ---------------------------------------------------------------------------
__global__ __launch_bounds__(128, 1)
void attn_kernel(const float* __restrict__ qf, const float* __restrict__ kf,
                 const float* __restrict__ vf, const float* __restrict__ pos,
                 const float* __restrict__ sigma_att, float* __restrict__ of) {
  extern __shared__ float smem[];
  float* kfs  = smem;           // 256*40
  float* vfs  = smem + 10240;   // 256*80
  float* poss = smem + 30720;   // 256*4
  float* pst  = smem + 31744;   // 4 waves * 16*16

  int bh = blockIdx.x;
  int b = bh >> 3, hh = bh & 7;
  int tidx = threadIdx.x;

  {  // stage kf tile (float4)
    const float4* src = (const float4*)(kf + (size_t)bh * (NBALL * QKDIM));
    float4* dst = (float4*)kfs;
    for (int idx = tidx; idx < 2560; idx += 128) dst[idx] = src[idx];
  }
  {  // stage vf tile with pad 72 -> 80
    const float4* src = (const float4*)(vf + (size_t)bh * (NBALL * VDIM));
    for (int idx = tidx; idx < 4608; idx += 128) {
      int row = idx / 18, col = idx % 18;
      *(float4*)(vfs + row * VPAD + col * 4) = src[idx];
    }
  }
  {  // stage positions (pad 3 -> 4)
    const float* src = pos + (size_t)b * (NBALL * 3);
    for (int idx = tidx; idx < 768; idx += 128) {
      int qq = idx / 3, d = idx % 3;
      poss[qq * 4 + d] = src[idx];
    }
  }
  float sigma = sigma_att[hh];
  __syncthreads();

  const float scale = 0.15811388300841898f;  // 1/sqrt(40)
  int wave = tidx >> 5, lane = tidx & 31;
  int hl = lane >> 4, lm = lane & 15;
  int koff = hl * 2;
  float* myp = pst + wave * 256;

#pragma unroll 1
  for (int qt = wave; qt < 16; qt += 4) {
    int q0 = qt * 16;

    // Q fragments
    v2f aq[10];
    const float* qrow = qf + ((size_t)bh * NBALL + q0 + lm) * QKDIM;
#pragma unroll
    for (int c = 0; c < 10; ++c) {
      aq[c].x = qrow[c * 4 + koff];
      aq[c].y = qrow[c * 4 + koff + 1];
    }
    if (qt + 4 < 16)
      __builtin_prefetch(qf + ((size_t)bh * NBALL + (qt + 4) * 16 + lm) * QKDIM, 0, 3);

    // ---- S = Q K^T * scale + sigma * dist ----
    float S[16][8];
#pragma unroll
    for (int kt = 0; kt < 16; ++kt) {
      v8f acc = v8f_zero();
#pragma unroll
      for (int c = 0; c < 10; ++c) {
        v2f bk;
        const float* kp = kfs + (kt * 16 + lm) * QKDIM + c * 4 + koff;
        bk.x = kp[0];
        bk.y = kp[1];
        acc = WMMA_F32(aq[c], bk, acc);
      }
#pragma unroll
      for (int vg = 0; vg < 8; ++vg) {
        int m = vg + hl * 8;
        int qq = q0 + m;
        int kk = kt * 16 + lm;
        float dx = poss[qq * 4 + 0] - poss[kk * 4 + 0];
        float dy = poss[qq * 4 + 1] - poss[kk * 4 + 1];
        float dz = poss[qq * 4 + 2] - poss[kk * 4 + 2];
        float dist = __builtin_amdgcn_sqrtf(dx * dx + dy * dy + dz * dz);
        S[kt][vg] = acc[vg] * scale + sigma * dist;
      }
    }

    // ---- row softmax (row M=vg+8*hl lives in one 16-lane half x 16 tiles) ----
    float rinv[8];
#pragma unroll
    for (int vg = 0; vg < 8; ++vg) {
      float mx = S[0][vg];
#pragma unroll
      for (int kt = 1; kt < 16; ++kt) mx = fmaxf(mx, S[kt][vg]);
      mx = fmaxf(mx, __shfl_xor(mx, 1, 32));
      mx = fmaxf(mx, __shfl_xor(mx, 2, 32));
      mx = fmaxf(mx, __shfl_xor(mx, 4, 32));
      mx = fmaxf(mx, __shfl_xor(mx, 8, 32));
      float sm = 0.f;
#pragma unroll
      for (int kt = 0; kt < 16; ++kt) {
        float p = __expf(S[kt][vg] - mx);
        S[kt][vg] = p;
        sm += p;
      }
      sm += __shfl_xor(sm, 1, 32);
      sm += __shfl_xor(sm, 2, 32);
      sm += __shfl_xor(sm, 4, 32);
      sm += __shfl_xor(sm, 8, 32);
      rinv[vg] = __builtin_amdgcn_rcpf(sm);
    }

    // ---- O = P V (transpose P through per-wave LDS tile; same-wave DS ops
    // are in-order so no barrier needed) ----
    v8f oacc[5];
#pragma unroll
    for (int ft = 0; ft < 5; ++ft) oacc[ft] = v8f_zero();

#pragma unroll
    for (int kt = 0; kt < 16; ++kt) {
#pragma unroll
      for (int vg = 0; vg < 8; ++vg) myp[(vg + hl * 8) * 16 + lm] = S[kt][vg];
#pragma unroll
      for (int c2 = 0; c2 < 4; ++c2) {
        v2f ap;
        const float* pp = myp + lm * 16 + c2 * 4 + koff;
        ap.x = pp[0];
        ap.y = pp[1];
#pragma unroll
        for (int ft = 0; ft < 5; ++ft) {
          v2f bv;
          const float* vp = vfs + (kt * 16 + c2 * 4 + koff) * VPAD + ft * 16 + lm;
          bv.x = vp[0];
          bv.y = vp[VPAD];
          oacc[ft] = WMMA_F32(ap, bv, oacc[ft]);
        }
      }
    }

    // ---- normalize by row sum and store ----
#pragma unroll
    for (int ft = 0; ft < 5; ++ft) {
      int feat = ft * 16 + lm;
#pragma unroll
      for (int vg = 0; vg < 8; ++vg) {
        if (feat < VDIM) {
          int m = vg + hl * 8;
          of[((size_t)bh * NBALL + q0 + m) * VDIM + feat] = oacc[ft][vg] * rinv[vg];
        }
      }
    }
  }
}

// ---------------------------------------------------------------------------
// Kernel C1 (WMMA): output multivector projection (32 in-ch -> 16 out-ch).
// One wave = one 16-point tile; single 16-wide n-tile of out-channels.
// A fragments are loaded on demand per contribution to cap VGPR pressure.
// grid = N/64 blocks x 128 threads.
// ---------------------------------------------------------------------------
__global__ __launch_bounds__(128, 1)
void out_mv_wmma(const float* __restrict__ of, const float* __restrict__ w_mv,
                 const float* __restrict__ w_s2mv, const float* __restrict__ b_mv,
                 float* __restrict__ out) {
  CONTRIB_TABLES
  int wv = threadIdx.x >> 5, lane = threadIdx.x & 31;
  int hl = lane >> 4, lm = lane & 15, koff = hl * 2;
  int n0 = (blockIdx.x * 4 + wv) * 16;

  int nl = n0 + lm;
  int bb = nl >> 8, qq = nl & 255;
  // Per-lane row base pointers for the 8 heads of `of`.
  const float* ofrow[8];
#pragma unroll
  for (int h = 0; h < 8; ++h)
    ofrow[h] = of + ((size_t)(bb * NHEAD + h) * NBALL + qq) * VDIM;

  int o = lm;
  const float* wbase = w_mv + (size_t)o * (32 * 9);  // + i*9 + k
  const float* w2 = w_s2mv + (size_t)o * 64;

  v8f acc[16];
#pragma unroll
  for (int a = 0; a < 16; ++a) acc[a] = v8f_zero();

#pragma unroll
  for (int ci = 0; ci < NCONTRIB; ++ci) {
    int a = CAa[ci], bc = CBb[ci], k = CKk[ci];
#pragma unroll
    for (int c = 0; c < 8; ++c) {  // K = 32 in-channels (i = h*4+m)
      int i0 = c * 4 + koff, i1 = i0 + 1;
      v2f xa;
      xa.x = ofrow[i0 >> 2][(i0 & 3) * 16 + bc];
      xa.y = ofrow[i1 >> 2][(i1 & 3) * 16 + bc];
      v2f wb;
      wb.x = wbase[i0 * 9 + k];
      wb.y = wbase[i1 * 9 + k];
      acc[a] = WMMA_F32(xa, wb, acc[a]);
    }
  }
  // scalar->mv into component 0 (K = 64, s = h*8+j)
#pragma unroll
  for (int c = 0; c < 16; ++c) {
    int s0 = c * 4 + koff, s1 = s0 + 1;
    v2f xa;
    xa.x = ofrow[s0 >> 3][64 + (s0 & 7)];
    xa.y = ofrow[s1 >> 3][64 + (s1 & 7)];
    v2f wb;
    wb.x = w2[s0];
    wb.y = w2[s1];
    acc[0] = WMMA_F32(xa, wb, acc[0]);
  }
  float bo = b_mv[o];
#pragma unroll
  for (int vg = 0; vg < 8; ++vg) acc[0][vg] += bo;

  // Store: out[n][o][a], 16 consecutive comps per (point, o) -> 4x float4.
#pragma unroll
  for (int vg = 0; vg < 8; ++vg) {
    int nn = n0 + vg + hl * 8;
    float* dst = out + (size_t)nn * 256 + o * 16;
#pragma unroll
    for (int c4 = 0; c4 < 4; ++c4) {
      float4 v;
      v.x = acc[c4 * 4 + 0][vg];
      v.y = acc[c4 * 4 + 1][vg];
      v.z = acc[c4 * 4 + 2][vg];
      v.w = acc[c4 * 4 + 3][vg];
      ((float4*)dst)[c4] = v;
    }
  }
}

// ---------------------------------------------------------------------------
// Kernel C2: output scalar projection (small, stays VALU).
// ---------------------------------------------------------------------------
__global__ __launch_bounds__(256)
void out_s_kernel(const float* __restrict__ of, const float* __restrict__ w_s,
                  const float* __restrict__ w_mv2s, const float* __restrict__ b_s,
                  float* __restrict__ out_s) {
  __shared__ float sw[32 * 64];
  __shared__ float s2[32 * 32];
  __shared__ float sb[32];
  for (int idx = threadIdx.x; idx < 32 * 64; idx += 256) sw[idx] = w_s[idx];
  for (int idx = threadIdx.x; idx < 32 * 32; idx += 256) s2[idx] = w_mv2s[idx];
  if (threadIdx.x < 32) sb[threadIdx.x] = b_s[threadIdx.x];
  __syncthreads();

  int n = blockIdx.x * 256 + threadIdx.x;
  int b = n >> 8, q = n & 255;

  float acc[32];
#pragma unroll
  for (int o = 0; o < 32; ++o) acc[o] = sb[o];

#pragma unroll 4
  for (int s = 0; s < 64; ++s) {
    int hh = s >> 3, j = s & 7;
    float xs = of[((size_t)(b * NHEAD + hh) * NBALL + q) * VDIM + 64 + j];
#pragma unroll
    for (int o = 0; o < 32; ++o) acc[o] += sw[o * 64 + s] * xs;
  }
#pragma unroll 4
  for (int c = 0; c < 32; ++c) {
    int hh = c >> 2, m = c & 3;
    float xc = of[((size_t)(b * NHEAD + hh) * NBALL + q) * VDIM + m * 16];  // comp 0
#pragma unroll
    for (int o = 0; o < 32; ++o) acc[o] += s2[o * 32 + c] * xc;
  }

  float* dst = out_s + (size_t)n * 32;
#pragma unroll
  for (int o = 0; o < 32; ++o) dst[o] = acc[o];
}

// ---------------------------------------------------------------------------
extern "C" void kernel_launch(void* const* d_in, const int* in_sizes, int n_in,
                              void* d_out, int out_size, void* d_ws, size_t ws_size,
                              hipStream_t stream) {
  (void)in_sizes; (void)n_in; (void)out_size; (void)ws_size;
  const float* x_mv       = (const float*)d_in[0];
  const float* x_s        = (const float*)d_in[1];
  const float* pos        = (const float*)d_in[3];   // pos_cartesian
  const float* w_qkv_mv   = (const float*)d_in[5];
  const float* w_qkv_s2mv = (const float*)d_in[6];
  const float* b_qkv_mv   = (const float*)d_in[7];
  const float* w_qkv_s    = (const float*)d_in[8];
  const float* w_qkv_mv2s = (const float*)d_in[9];
  const float* b_qkv_s    = (const float*)d_in[10];
  const float* w_out_mv   = (const float*)d_in[11];
  const float* w_out_s2mv = (const float*)d_in[12];
  const float* b_out_mv   = (const float*)d_in[13];
  const float* w_out_s    = (const float*)d_in[14];
  const float* w_out_mv2s = (const float*)d_in[15];
  const float* b_out_s    = (const float*)d_in[16];
  const float* sigma_att  = (const float*)d_in[17];

  float* ws = (float*)d_ws;
  float* qf = ws;
  float* kf = qf + QF_ELEMS;
  float* vf = kf + QF_ELEMS;
  float* of = vf + VF_ELEMS;

  float* out_mv = (float*)d_out;
  float* out_s  = out_mv + (size_t)NPTS * 256;

  qkv_mv_wmma<<<NPTS / 64, 128, 0, stream>>>(
      x_mv, x_s, w_qkv_mv, w_qkv_s2mv, b_qkv_mv, qf, kf, vf);
  qkv_s_kernel<<<(NPTS * 24) / 256, 256, 0, stream>>>(
      x_mv, x_s, w_qkv_s, w_qkv_mv2s, b_qkv_s, qf, kf, vf);
  attn_kernel<<<2048, 128, 131072, stream>>>(qf, kf, vf, pos, sigma_att, of);
  out_mv_wmma<<<NPTS / 64, 128, 0, stream>>>(
      of, w_out_mv, w_out_s2mv, b_out_mv, out_mv);
  out_s_kernel<<<NPTS / 256, 256, 0, stream>>>(
      of, w_out_s, w_out_mv2s, b_out_s, out_s);
}